// ModifiedMistralAttention_73787538145865
// MI455X (gfx1250) — compile-verified
//
#include <hip/hip_runtime.h>

typedef float v2f __attribute__((ext_vector_type(2)));
typedef float v8f __attribute__((ext_vector_type(8)));

#define B_  2
#define S_  96
#define H_  96
#define D_  96
#define NP  (B_ * S_)          // 192 positions
#define SCALE_ 0.10206207261596575f   // 1/sqrt(96)

// ---------------------------------------------------------------------------
// Kernel 1: per position p, t[i] = sum_k Q[p, idx, k] * V[p, k, i]
//   Q layout (B,S,H,D): row Q[p*H*D + idx*D + k]
//   V layout (B,S,H,D): V[p*H*D + k*D + i]  -> i coalesced across lanes
// ---------------------------------------------------------------------------
__global__ void k_tvec(const float* __restrict__ Q,
                       const float* __restrict__ V,
                       const int*   __restrict__ idxp,
                       float*       __restrict__ T) {
    __shared__ float qsh[D_];
    const int p = blockIdx.x;        // 0..191
    const int i = threadIdx.x;       // 0..95
    const int idx = *idxp;

    const float* __restrict__ qrow = Q + (size_t)p * H_ * D_ + (size_t)idx * D_;
    if (i < D_) qsh[i] = qrow[i];
    __syncthreads();

    const float* __restrict__ vbase = V + (size_t)p * H_ * D_;
    float acc = 0.0f;
#pragma unroll 8
    for (int k = 0; k < H_; ++k) {
        acc = fmaf(qsh[k], vbase[(size_t)k * D_ + i], acc);
    }
    T[(size_t)p * D_ + i] = acc;
}

// ---------------------------------------------------------------------------
// Kernel 2: U(192x96) = T(192x96) * W(96x96), W = depth_weights[idx]
// fp32 WMMA 16x16x4, one wave per 16x16 output tile, 24 K-steps.
//   A 16x4 f32: 2 VGPRs/lane -> v2f; lanes 0-15: K=0,1; lanes 16-31: K=2,3
//   B 4x16 f32: 2 VGPRs/lane -> v2f; VGPR0 rows K=0/K=2, VGPR1 rows K=1/K=3
//   C/D 16x16 f32: v8f; element j -> M = j + 8*(lane>=16), N = lane&15
// ---------------------------------------------------------------------------
__global__ void k_gemm_wmma(const float* __restrict__ T,
                            const float* __restrict__ DW,
                            const int*   __restrict__ idxp,
                            float*       __restrict__ U) {
    const int tile = blockIdx.x;          // 0..71
    const int tn   = tile % (D_ / 16);    // col tile 0..5
    const int tm   = tile / (D_ / 16);    // row tile 0..11
    const int lane = threadIdx.x & 31;
    const int half = lane >> 4;           // 0 or 1
    const int lid  = lane & 15;

    const float* __restrict__ W = DW + (size_t)(*idxp) * D_ * D_;
    const int row0 = tm * 16;
    const int col0 = tn * 16;

    v8f c = {};
    for (int kk = 0; kk < D_; kk += 4) {
        const int ka = kk + 2 * half;     // this lane's K base (0/2 offset)
        v2f a, b;
        a.x = T[(size_t)(row0 + lid) * D_ + ka];
        a.y = T[(size_t)(row0 + lid) * D_ + ka + 1];
        b.x = W[(size_t)ka       * D_ + col0 + lid];
        b.y = W[(size_t)(ka + 1) * D_ + col0 + lid];
        // (neg_a, A, neg_b, B, c_mod, C, reuse_a, reuse_b)
        c = __builtin_amdgcn_wmma_f32_16x16x4_f32(false, a, false, b,
                                                  (short)0, c, false, false);
    }

#pragma unroll
    for (int j = 0; j < 8; ++j) {
        U[(size_t)(row0 + j + 8 * half) * D_ + col0 + lid] = c[j];
    }
}

// ---------------------------------------------------------------------------
// Kernel 3: out[p, q, v] = scale * K[p, idx, q] * U[p, v]
// Pure streaming rank-1 expansion; v coalesced across lanes.
// ---------------------------------------------------------------------------
__global__ void k_outer(const float* __restrict__ Kmat,
                        const float* __restrict__ U,
                        const int*   __restrict__ idxp,
                        float*       __restrict__ out) {
    const int gid = blockIdx.x * blockDim.x + threadIdx.x;
    const int total = NP * D_ * D_;
    if (gid >= total) return;
    const int v = gid % D_;
    const int r = gid / D_;
    const int q = r % D_;
    const int p = r / D_;
    const float kv = Kmat[((size_t)p * H_ + (size_t)(*idxp)) * D_ + q];
    out[gid] = SCALE_ * kv * U[(size_t)p * D_ + v];
}

// ---------------------------------------------------------------------------
extern "C" void kernel_launch(void* const* d_in, const int* in_sizes, int n_in,
                              void* d_out, int out_size, void* d_ws, size_t ws_size,
                              hipStream_t stream) {
    const float* Q   = (const float*)d_in[0];
    const float* K   = (const float*)d_in[1];
    const float* V   = (const float*)d_in[2];
    const float* DW  = (const float*)d_in[3];  // depth_weights (8,96,96)
    const int*   idx = (const int*)  d_in[4];  // user_depth_index (scalar)
    float* out = (float*)d_out;                // (B, S, D*D) = 1,769,472 floats

    // workspace: T (192x96) then U (192x96), 147,456 bytes total
    float* T = (float*)d_ws;
    float* U = T + (size_t)NP * D_;

    k_tvec     <<<NP, 96, 0, stream>>>(Q, V, idx, T);
    k_gemm_wmma<<<(NP / 16) * (D_ / 16), 32, 0, stream>>>(T, DW, idx, U);

    const int total  = NP * D_ * D_;
    const int blocks = (total + 255) / 256;
    k_outer    <<<blocks, 256, 0, stream>>>(K, U, idx, out);
}